// Attention_47768626266740
// MI455X (gfx1250) — compile-verified
//
#include <hip/hip_runtime.h>
#include <hip/hip_bf16.h>
#include <math.h>

// Problem constants (from reference): B=128, T=2048, H=256
#define B_ 128
#define T_ 2048
#define H_ 256
#define S_ 8          // split-T factor for the streaming kernel
#define ROWS_PER_WG (T_ / S_)   // 256 t-rows per workgroup
#define ROWS_PER_WAVE (ROWS_PER_WG / 8) // 32 rows per wave (8 waves / block)

typedef __attribute__((ext_vector_type(2))) float v2f;
typedef __attribute__((ext_vector_type(8))) float v8f;

// ---------------------------------------------------------------------------
// Kernel 1: V[b,h] = sum_k h_t[b,k] * W_score[h,k]   (128x256x256 fp32 GEMM)
// D = A x Bmat with A = h_t (M=128 x K=256), Bmat[k,n] = W_score[n,k].
// Exact fp32 via V_WMMA_F32_16X16X4_F32. grid = (8 M-tiles, 16 N-tiles), 1 wave.
// ---------------------------------------------------------------------------
__global__ __launch_bounds__(32) void k_vmat(const float* __restrict__ hid,
                                             const float* __restrict__ Wscore,
                                             float* __restrict__ V) {
  const int tm   = blockIdx.x;          // batch tile (M)
  const int tn   = blockIdx.y;          // h tile (N)
  const int lane = threadIdx.x;
  const int half = lane >> 4;           // k sub-half per ISA A/B layout
  const int r    = lane & 15;           // M index for A frag, N index for B frag
  const size_t hstride = (size_t)T_ * H_;
  const float* ht = hid + (size_t)(T_ - 1) * H_;   // h_t rows, stride T*H

  v8f acc = {};
  for (int k0 = 0; k0 < H_; k0 += 4) {
    const int k = k0 + 2 * half;
    // A frag: a[vg] = h_t[tm*16+r][k+vg]  (contiguous pair)
    const float2 av = *(const float2*)(ht + (size_t)(tm * 16 + r) * hstride + k);
    // B frag: b[vg] = W_score[tn*16+r][k+vg] (contiguous pair; Bmat[k][n]=W_score[n][k])
    const float2 bv = *(const float2*)(Wscore + (size_t)(tn * 16 + r) * H_ + k);
    v2f a; a[0] = av.x; a[1] = av.y;
    v2f b; b[0] = bv.x; b[1] = bv.y;
    acc = __builtin_amdgcn_wmma_f32_16x16x4_f32(false, a, false, b,
                                                (short)0, acc, false, false);
  }
#pragma unroll
  for (int j = 0; j < 8; ++j) {
    const int m = tm * 16 + j + 8 * half;
    const int n = tn * 16 + r;
    V[(size_t)m * H_ + n] = acc[j];
  }
}

// ---------------------------------------------------------------------------
// Kernel 2: streaming pass over hidden (the 256 MB / bandwidth-bound part).
// grid = (B=128, S=8), 256 threads (8 waves). Each wave handles 32 t-rows of
// one batch; lane owns 8 contiguous h-floats. Online softmax fused with the
// context accumulation (flash-style): single read of hidden_states.
// Emits per-(b,split) partials: running max m, rescaled sum, rescaled c[256].
// ---------------------------------------------------------------------------
__global__ __launch_bounds__(256) void k_stream(const float* __restrict__ hid,
                                                const float* __restrict__ V,
                                                float* __restrict__ pm,
                                                float* __restrict__ psum,
                                                float* __restrict__ pc) {
  const int b    = blockIdx.x;
  const int s    = blockIdx.y;
  const int wave = threadIdx.x >> 5;
  const int lane = threadIdx.x & 31;

  // v[b] fragment: 8 floats per lane
  const float* vrow = V + (size_t)b * H_ + lane * 8;
  const float4 v0 = *(const float4*)vrow;
  const float4 v1 = *(const float4*)(vrow + 4);

  float m = -INFINITY, run = 0.f;
  float c[8] = {0.f, 0.f, 0.f, 0.f, 0.f, 0.f, 0.f, 0.f};

  const int t0 = s * ROWS_PER_WG + wave * ROWS_PER_WAVE;
  const float* base = hid + ((size_t)b * T_ + t0) * H_ + lane * 8;

  for (int i = 0; i < ROWS_PER_WAVE; ++i) {
    const float* row = base + (size_t)i * H_;
    const float4 h0 = *(const float4*)row;
    const float4 h1 = *(const float4*)(row + 4);
    // partial dot
    float p = h0.x * v0.x + h0.y * v0.y + h0.z * v0.z + h0.w * v0.w
            + h1.x * v1.x + h1.y * v1.y + h1.z * v1.z + h1.w * v1.w;
    // wave32 butterfly reduction -> full score s_t in every lane
    p += __shfl_xor(p, 16);
    p += __shfl_xor(p, 8);
    p += __shfl_xor(p, 4);
    p += __shfl_xor(p, 2);
    p += __shfl_xor(p, 1);
    // online softmax update
    const float mn = fmaxf(m, p);
    const float sc = __expf(m - mn);   // 0 on first iter (m=-inf), else rescale
    const float pe = __expf(p - mn);
    run = run * sc + pe;
    m = mn;
    c[0] = c[0] * sc + pe * h0.x;
    c[1] = c[1] * sc + pe * h0.y;
    c[2] = c[2] * sc + pe * h0.z;
    c[3] = c[3] * sc + pe * h0.w;
    c[4] = c[4] * sc + pe * h1.x;
    c[5] = c[5] * sc + pe * h1.y;
    c[6] = c[6] * sc + pe * h1.z;
    c[7] = c[7] * sc + pe * h1.w;
  }

  // cross-wave combine in LDS
  __shared__ float sm[8];
  __shared__ float ss[8];
  __shared__ float scx[8][H_];
#pragma unroll
  for (int j = 0; j < 8; ++j) scx[wave][lane * 8 + j] = c[j];
  if (lane == 0) { sm[wave] = m; ss[wave] = run; }
  __syncthreads();

  const int h = threadIdx.x;                 // 0..255
  float M = sm[0];
#pragma unroll
  for (int w = 1; w < 8; ++w) M = fmaxf(M, sm[w]);
  float tot = 0.f, ch = 0.f;
#pragma unroll
  for (int w = 0; w < 8; ++w) {
    const float e = __expf(sm[w] - M);
    tot += ss[w] * e;
    ch  += scx[w][h] * e;
  }
  const int idx = b * S_ + s;
  pc[(size_t)idx * H_ + h] = ch;
  if (threadIdx.x == 0) { pm[idx] = M; psum[idx] = tot; }
}

// ---------------------------------------------------------------------------
// Kernel 3: combine split-T partials -> context; build pre_activation (B,512)
// ---------------------------------------------------------------------------
__global__ __launch_bounds__(256) void k_combine(const float* __restrict__ hid,
                                                 const float* __restrict__ pm,
                                                 const float* __restrict__ psum,
                                                 const float* __restrict__ pc,
                                                 float* __restrict__ preact) {
  const int b = blockIdx.x;
  const int h = threadIdx.x;
  float M = pm[b * S_];
#pragma unroll
  for (int s = 1; s < S_; ++s) M = fmaxf(M, pm[b * S_ + s]);
  float tot = 0.f, ch = 0.f;
#pragma unroll
  for (int s = 0; s < S_; ++s) {
    const float e = __expf(pm[b * S_ + s] - M);
    tot += psum[b * S_ + s] * e;
    ch  += pc[(size_t)(b * S_ + s) * H_ + h] * e;
  }
  preact[(size_t)b * 512 + h]       = ch / tot;                          // context
  preact[(size_t)b * 512 + 256 + h] = hid[((size_t)b * T_ + (T_ - 1)) * H_ + h]; // h_t
}

// ---------------------------------------------------------------------------
// Kernel 4: out = tanh(pre_activation (128x512) @ W_att (512x128)), fp32 WMMA.
// grid = (8,8) 16x16 tiles, 1 wave each, K=512 in steps of 4.
// ---------------------------------------------------------------------------
__global__ __launch_bounds__(32) void k_att(const float* __restrict__ preact,
                                            const float* __restrict__ Watt,
                                            float* __restrict__ out) {
  const int tm   = blockIdx.x;
  const int tn   = blockIdx.y;
  const int lane = threadIdx.x;
  const int half = lane >> 4;
  const int r    = lane & 15;

  v8f acc = {};
  for (int k0 = 0; k0 < 512; k0 += 4) {
    const int k = k0 + 2 * half;
    const float2 av = *(const float2*)(preact + (size_t)(tm * 16 + r) * 512 + k);
    v2f a; a[0] = av.x; a[1] = av.y;
    v2f b;
    b[0] = Watt[(size_t)k * 128 + tn * 16 + r];
    b[1] = Watt[(size_t)(k + 1) * 128 + tn * 16 + r];
    acc = __builtin_amdgcn_wmma_f32_16x16x4_f32(false, a, false, b,
                                                (short)0, acc, false, false);
  }
#pragma unroll
  for (int j = 0; j < 8; ++j) {
    const int m = tm * 16 + j + 8 * half;
    const int n = tn * 16 + r;
    out[(size_t)m * 128 + n] = tanhf(acc[j]);
  }
}

// ---------------------------------------------------------------------------
extern "C" void kernel_launch(void* const* d_in, const int* in_sizes, int n_in,
                              void* d_out, int out_size, void* d_ws, size_t ws_size,
                              hipStream_t stream) {
  const float* hid    = (const float*)d_in[0];   // (128, 2048, 256) f32
  const float* Wscore = (const float*)d_in[1];   // (256, 256) f32
  const float* Watt   = (const float*)d_in[2];   // (512, 128) f32
  float* out = (float*)d_out;                    // (128, 128) f32
  float* ws  = (float*)d_ws;

  // workspace layout (floats): V | pm | psum | pc | preact  (~1.45 MB total)
  float* V      = ws;                       // 128*256          = 32768
  float* pm     = ws + 32768;               // 128*8            = 1024
  float* psum   = ws + 33792;               // 128*8            = 1024
  float* pc     = ws + 34816;               // 128*8*256        = 262144
  float* preact = ws + 296960;              // 128*512          = 65536

  k_vmat   <<<dim3(8, 16),  32,  0, stream>>>(hid, Wscore, V);
  k_stream <<<dim3(B_, S_), 256, 0, stream>>>(hid, V, pm, psum, pc);
  k_combine<<<B_,           256, 0, stream>>>(hid, pm, psum, pc, preact);
  k_att    <<<dim3(8, 8),   32,  0, stream>>>(preact, Watt, out);
}